// MyEmbedding_41798621725208
// MI455X (gfx1250) — compile-verified
//
#include <hip/hip_runtime.h>
#include <cstdint>

// MyEmbedding gather for MI455X (gfx1250).
//
// Pure-bandwidth kernel: ~240 MB of HBM traffic, 0 FLOPs -> ~10us roofline at
// 23.3 TB/s. Strategy:
//   * 2-D grid (L, B): one 256-thread block per token row (8 KiB), no integer
//     division anywhere (b = blockIdx.y)
//   * uniform-per-block source-pointer select (static tables vs image_embedding),
//     index fetched via scalar load (block-uniform)
//   * CDNA5 async DMA path: global_load_async_to_lds_b128 then
//     global_store_async_from_lds_b128 (ASYNCcnt), so row data never occupies
//     VGPRs -> minimal register pressure, maximal wave occupancy
//   * TH_STORE_NT on the streamed 128 MiB output so it does not evict the
//     ~105 MB of distinct gathered rows that fit in the 192 MB L2

#define VW        32000   // num_embeddings
#define V_STATIC  32008   // VW + 4 special tables * 2 rows
#define EMB_D     2048    // embedding_dim (floats)
#define T_VIS     256     // perceiver latents per batch
#define ROW_BYTES (EMB_D * 4)   // 8192 bytes per row

__global__ __launch_bounds__(256)
void emb_gather_async_kernel(const float* __restrict__ weight,
                             const float* __restrict__ img_tok,
                             const float* __restrict__ pet_tok,
                             const float* __restrict__ reg_tok,
                             const float* __restrict__ tpl_tok,
                             const float* __restrict__ img_emb,   // (B, T_VIS, D)
                             const int*   __restrict__ text,      // (B, L) flat
                             float* __restrict__ out)             // (B, L, D)
{
    __shared__ __align__(16) char buf[ROW_BYTES];

    const int l     = (int)blockIdx.x;           // position in sequence
    const int b     = (int)blockIdx.y;           // batch
    const int token = b * (int)gridDim.x + l;    // flat (B, L) index
    const int t     = (int)threadIdx.x;          // 0..255

    const int idx = text[token];                 // uniform within the block

    // Select source row pointer (reference concatenates 5 tables then vis).
    const char* src;
    if (idx < VW) {
        src = (const char*)(weight + (size_t)idx * EMB_D);
    } else if (idx < VW + 2) {
        src = (const char*)(img_tok + (size_t)(idx - VW) * EMB_D);
    } else if (idx < VW + 4) {
        src = (const char*)(pet_tok + (size_t)(idx - (VW + 2)) * EMB_D);
    } else if (idx < VW + 6) {
        src = (const char*)(reg_tok + (size_t)(idx - (VW + 4)) * EMB_D);
    } else if (idx < V_STATIC) {
        src = (const char*)(tpl_tok + (size_t)(idx - (VW + 6)) * EMB_D);
    } else {
        int v = idx - V_STATIC;
        v = v < 0 ? 0 : (v > (T_VIS - 1) ? (T_VIS - 1) : v);  // reference clip
        src = (const char*)(img_emb + ((size_t)b * T_VIS + v) * EMB_D);
    }

    // ISA 10.2: low 32 bits of a flat LDS pointer are the LDS byte offset.
    const uint32_t lbase = (uint32_t)(uintptr_t)buf;
    const char*    dst   = (const char*)out + (size_t)token * ROW_BYTES;

    // Each thread DMAs two disjoint 16-byte chunks: [t*16] and [t*16 + 4096].
    // HBM -> LDS (async, ASYNCcnt). No VGPR data movement.
#pragma unroll
    for (int c = 0; c < 2; ++c) {
        const uint32_t off = (uint32_t)(t * 16 + c * 4096);
        const uint64_t ga  = (uint64_t)(uintptr_t)(src + off);
        const uint32_t la  = lbase + off;
        asm volatile("global_load_async_to_lds_b128 %0, %1, off"
                     :: "v"(la), "v"(ga) : "memory");
    }

    // Wait for this wave's async loads; each lane only reads back LDS bytes it
    // wrote itself, so no cross-wave barrier is required.
    asm volatile("s_wait_asynccnt 0" ::: "memory");

    // LDS -> HBM (async), non-temporal so the output stream does not thrash L2.
#pragma unroll
    for (int c = 0; c < 2; ++c) {
        const uint32_t off = (uint32_t)(t * 16 + c * 4096);
        const uint64_t ga  = (uint64_t)(uintptr_t)(dst + off);
        const uint32_t la  = lbase + off;
        asm volatile("global_store_async_from_lds_b128 %0, %1, off th:TH_STORE_NT"
                     :: "v"(ga), "v"(la) : "memory");
    }

    // Drain stores before the block retires (S_ENDPGM also waits idle).
    asm volatile("s_wait_asynccnt 0" ::: "memory");
}

extern "C" void kernel_launch(void* const* d_in, const int* in_sizes, int n_in,
                              void* d_out, int out_size, void* d_ws, size_t ws_size,
                              hipStream_t stream) {
    const float* weight  = (const float*)d_in[0];  // (V, D)
    const float* img_tok = (const float*)d_in[1];  // (2, D)
    const float* pet_tok = (const float*)d_in[2];  // (2, D)
    const float* reg_tok = (const float*)d_in[3];  // (2, D)
    const float* tpl_tok = (const float*)d_in[4];  // (2, D)
    const float* img_emb = (const float*)d_in[5];  // (B, T_VIS, D)
    const int*   text    = (const int*)d_in[6];    // (B, L)
    float*       out     = (float*)d_out;          // (B, L, D)

    const int BL = in_sizes[6];                    // B * L tokens
    int B = in_sizes[5] / (T_VIS * EMB_D);         // batch
    if (B < 1) B = 1;
    const int L = BL / B;

    dim3 grid((unsigned)L, (unsigned)B);
    dim3 block(256);
    emb_gather_async_kernel<<<grid, block, 0, stream>>>(
        weight, img_tok, pet_tok, reg_tok, tpl_tok, img_emb, text, out);
}